// GCN_19722489823529
// MI455X (gfx1250) — compile-verified
//
#include <hip/hip_runtime.h>
#include <math.h>

typedef float v2f __attribute__((ext_vector_type(2)));
typedef float v8f __attribute__((ext_vector_type(8)));

#define D_IN  50
#define D_HID 32
#define D_OUT 5

// ---------------- utility kernels ----------------

__global__ void gcn_zero_kernel(float* __restrict__ p, int n) {
    int i = blockIdx.x * blockDim.x + threadIdx.x;
    if (i < n) p[i] = 0.0f;
}

__global__ void gcn_deg_kernel(const int* __restrict__ dst, float* __restrict__ deg, int E) {
    int e = blockIdx.x * blockDim.x + threadIdx.x;
    if (e < E) atomicAdd(&deg[dst[e]], 1.0f);
}

// ---------------- WMMA GEMM: Y[M x LDY] = X[M x K] @ W[K x Wn] (+ bias) ----------------
// One wave computes one 16x16 tile of Y via V_WMMA_F32_16X16X4_F32.
// W is staged in LDS, zero-padded to KP x (NTN*16) and pair-interleaved so each
// B fragment is a single aligned ds_load_b64; A fragments are single global_load_b64.
// Requires M % 16 == 0 (true here: 100000 = 6250*16).
template <int K, int Wn, int LDY, int NTN, bool BIAS>
__global__ void __launch_bounds__(256)
gcn_wmma_gemm(const float* __restrict__ X, const float* __restrict__ W,
              const float* __restrict__ bias, float* __restrict__ Y,
              int total_tiles) {
    constexpr int KP  = (K + 3) & ~3;      // K padded to multiple of 4
    constexpr int LDW = NTN * 16;          // padded column count
    __shared__ float wlds[KP / 2][LDW][2]; // [k>>1][col][k&1] : (k,k+1) contiguous

    // cooperative zero-padded fill of W into LDS
    for (int idx = threadIdx.x; idx < KP * LDW; idx += blockDim.x) {
        int r = idx / LDW;                 // k index
        int c = idx % LDW;                 // column
        float v = (r < K && c < Wn) ? W[r * Wn + c] : 0.0f;
        wlds[r >> 1][c][r & 1] = v;
    }
    __syncthreads();

    int wave = threadIdx.x >> 5;
    int lane = threadIdx.x & 31;
    int t = blockIdx.x * 8 + wave;
    if (t >= total_tiles) return;          // wave-uniform; EXEC all-ones inside

    int m0   = (t / NTN) * 16;
    int n0   = (t % NTN) * 16;
    int half = lane >> 4;                  // selects K pair {0,1} vs {2,3} within chunk
    int l    = lane & 15;
    const float* xrow = X + (size_t)(m0 + l) * K;  // lane's A row
    int coll = n0 + l;                     // lane's B/D column

    v8f c = {};
    constexpr int K4 = K & ~3;
#pragma unroll
    for (int k0 = 0; k0 < K4; k0 += 4) {
        int ka = k0 + half * 2;
        v2f a = *reinterpret_cast<const v2f*>(xrow + ka);
        v2f b = *reinterpret_cast<const v2f*>(&wlds[ka >> 1][coll][0]);
        c = __builtin_amdgcn_wmma_f32_16x16x4_f32(false, a, false, b,
                                                  (short)0, c, false, false);
    }
    if constexpr ((K & 3) != 0) {          // tail chunk (K=50): B rows >= K are zero
        int ka = K4 + half * 2;
        const float* ap = (ka < K) ? (xrow + ka) : xrow;  // safe addr; value x0 via B=0
        v2f a = *reinterpret_cast<const v2f*>(ap);
        v2f b = *reinterpret_cast<const v2f*>(&wlds[ka >> 1][coll][0]);
        c = __builtin_amdgcn_wmma_f32_16x16x4_f32(false, a, false, b,
                                                  (short)0, c, false, false);
    }

    float bv = 0.0f;
    if constexpr (BIAS) bv = (coll < Wn) ? bias[coll] : 0.0f;  // padded cols: exact 0
    float* yp = Y + (size_t)(m0 + half * 8) * LDY + coll;
#pragma unroll
    for (int r = 0; r < 8; ++r) yp[r * LDY] = c[r] + bv;
}

// ---------------- edge scatter kernels ----------------

// layer 1: acc1[dst] += w[e] * xr1[src], 32 floats/edge, one thread per (edge, 4-col group)
__global__ void gcn_scatter_w32(const int* __restrict__ src, const int* __restrict__ dst,
                                const float* __restrict__ w, const float* __restrict__ xr,
                                float* __restrict__ acc, int E) {
    int gid = blockIdx.x * blockDim.x + threadIdx.x;
    int e = gid >> 3;
    if (e >= E) return;
    int cg = (gid & 7) * 4;
    int s = src[e], d = dst[e];
    float ww = w[e];
    const float4 v = *reinterpret_cast<const float4*>(xr + (size_t)s * D_HID + cg);
    float* p = acc + (size_t)d * D_HID + cg;
    atomicAdd(p + 0, v.x * ww);
    atomicAdd(p + 1, v.y * ww);
    atomicAdd(p + 2, v.z * ww);
    atomicAdd(p + 3, v.w * ww);
}

// layer 2: acc2[dst] += hr2[src], 5 floats/edge (hr2 stride 16, acc2 stride 8)
__global__ void gcn_scatter_5(const int* __restrict__ src, const int* __restrict__ dst,
                              const float* __restrict__ hr, float* __restrict__ acc, int E) {
    int e = blockIdx.x * blockDim.x + threadIdx.x;
    if (e >= E) return;
    int s = src[e], d = dst[e];
    const float4 v = *reinterpret_cast<const float4*>(hr + (size_t)s * 16);
    float v4 = hr[(size_t)s * 16 + 4];
    float* p = acc + (size_t)d * 8;
    atomicAdd(p + 0, v.x);
    atomicAdd(p + 1, v.y);
    atomicAdd(p + 2, v.z);
    atomicAdd(p + 3, v.w);
    atomicAdd(p + 4, v4);
}

// ---------------- fused finalize kernels ----------------

// h = relu(acc1/max(deg,1) + xo1), elementwise over N*32
__global__ void gcn_finalize1(const float* __restrict__ acc, const float* __restrict__ xo,
                              const float* __restrict__ deg, float* __restrict__ h, int total) {
    int i = blockIdx.x * blockDim.x + threadIdx.x;
    if (i >= total) return;
    int node = i >> 5;                      // /32
    float dg = fmaxf(deg[node], 1.0f);
    float v = acc[i] / dg + xo[i];
    h[i] = fmaxf(v, 0.0f);
}

// out = log_softmax(acc2/max(deg,1) + ho2) over 5 logits, one thread per node
__global__ void gcn_finalize2(const float* __restrict__ acc2, const float* __restrict__ ho,
                              const float* __restrict__ deg, float* __restrict__ out, int N) {
    int i = blockIdx.x * blockDim.x + threadIdx.x;
    if (i >= N) return;
    float dg = fmaxf(deg[i], 1.0f);
    float z[D_OUT];
    float m = -INFINITY;
#pragma unroll
    for (int c = 0; c < D_OUT; ++c) {
        z[c] = acc2[(size_t)i * 8 + c] / dg + ho[(size_t)i * 16 + c];
        m = fmaxf(m, z[c]);
    }
    float s = 0.0f;
#pragma unroll
    for (int c = 0; c < D_OUT; ++c) s += expf(z[c] - m);
    float ls = logf(s);
#pragma unroll
    for (int c = 0; c < D_OUT; ++c) out[(size_t)i * D_OUT + c] = z[c] - m - ls;
}

// ---------------- launcher ----------------

extern "C" void kernel_launch(void* const* d_in, const int* in_sizes, int n_in,
                              void* d_out, int out_size, void* d_ws, size_t ws_size,
                              hipStream_t stream) {
    const float* x      = (const float*)d_in[0];
    const int*   ei     = (const int*)  d_in[1];
    const float* w      = (const float*)d_in[2];
    const float* Wrel1  = (const float*)d_in[3];
    const float* Wroot1 = (const float*)d_in[4];
    const float* b1     = (const float*)d_in[5];
    const float* Wrel2  = (const float*)d_in[6];
    const float* Wroot2 = (const float*)d_in[7];
    const float* b2     = (const float*)d_in[8];
    float* out = (float*)d_out;

    const int N = in_sizes[0] / D_IN;   // 100000 (divisible by 16)
    const int E = in_sizes[2];          // 1600000
    const int* src = ei;                // edge_index row 0
    const int* dst = ei + E;            // edge_index row 1

    // workspace layout (floats)
    float* ws   = (float*)d_ws;
    float* xr1  = ws;                        // N*32  : x @ W_rel1
    float* xo1  = xr1  + (size_t)N * 32;     // N*32  : x @ W_root1 + b1
    float* h    = xo1  + (size_t)N * 32;     // N*32  : relu(layer1)
    float* hr2  = h    + (size_t)N * 32;     // N*16  : h @ W_rel2 (cols 0..4 valid)
    float* ho2  = hr2  + (size_t)N * 16;     // N*16  : h @ W_root2 + b2
    float* deg  = ho2  + (size_t)N * 16;     // N     : in-degree
    float* acc1 = deg  + (size_t)N;          // N*32  : scatter accumulator L1
    float* acc2 = acc1 + (size_t)N * 32;     // N*8   : scatter accumulator L2
    const int zn = N * (1 + 32 + 8);         // deg+acc1+acc2 contiguous

    const int TB = 256;
    gcn_zero_kernel<<<(zn + TB - 1) / TB, TB, 0, stream>>>(deg, zn);
    gcn_deg_kernel <<<(E  + TB - 1) / TB, TB, 0, stream>>>(dst, deg, E);

    const int mt = N / 16;                   // 6250 row tiles

    // layer 1 projections (WMMA fp32): 2 column tiles
    {
        const int total = mt * 2;
        const int blocks = (total + 7) / 8;
        gcn_wmma_gemm<D_IN, D_HID, 32, 2, false>
            <<<blocks, TB, 0, stream>>>(x, Wrel1,  nullptr, xr1, total);
        gcn_wmma_gemm<D_IN, D_HID, 32, 2, true>
            <<<blocks, TB, 0, stream>>>(x, Wroot1, b1,      xo1, total);
    }

    gcn_scatter_w32<<<(E * 8 + TB - 1) / TB, TB, 0, stream>>>(src, dst, w, xr1, acc1, E);
    gcn_finalize1  <<<((size_t)N * 32 + TB - 1) / TB, TB, 0, stream>>>(acc1, xo1, deg, h, N * 32);

    // layer 2 projections (WMMA fp32): one 16-wide padded column tile
    {
        const int total = mt;
        const int blocks = (total + 7) / 8;
        gcn_wmma_gemm<D_HID, D_OUT, 16, 1, false>
            <<<blocks, TB, 0, stream>>>(h, Wrel2,  nullptr, hr2, total);
        gcn_wmma_gemm<D_HID, D_OUT, 16, 1, true>
            <<<blocks, TB, 0, stream>>>(h, Wroot2, b2,      ho2, total);
    }

    gcn_scatter_5<<<(E + TB - 1) / TB, TB, 0, stream>>>(src, dst, hr2, acc2, E);
    gcn_finalize2<<<(N + TB - 1) / TB, TB, 0, stream>>>(acc2, ho2, deg, out, N);
}